// HNetLoss_20822001451157
// MI455X (gfx1250) — compile-verified
//
#include <hip/hip_runtime.h>
#include <hip/hip_bf16.h>

typedef __attribute__((ext_vector_type(2))) float v2f;
typedef __attribute__((ext_vector_type(8))) float v8f;

#define NLANES   5
#define EPS_DEN  1e-5f
#define RIDGE    1e-6f
#define NB       8
#define HH       256
#define WW       512
#define NPIX     (HH * WW)          // 131072

// workspace layout (floats)
#define ACC_OFF    0                 // [8][5][24] : rows i=0..3, cols j=0..5 (j<4: A, j=4: rhs, j=5: sum m*|den|)
#define ACC_STRIDE 24
#define ACC_TOTAL  (NB * NLANES * ACC_STRIDE)   // 960
#define SSE_OFF    960               // [8][5]
#define W_OFF      1000              // [8][5][4]
#define WS_FLOATS  1200

__global__ void k_zero(float* ws) {
    int i = blockIdx.x * blockDim.x + threadIdx.x;
    if (i < WS_FLOATS) ws[i] = 0.0f;
}

// Pass 1: masked normal equations via V_WMMA_F32_16X16X4_F32.
// grid = 256 blocks x 256 threads (8 waves). Block -> one 4096-pixel slice of one batch.
// Per 4-pixel chunk, two f32 WMMAs accumulate
//   D[(l,i), j] += sum_t  m_{l,t} * yp_t^(3-i) * col_j(t),  col = [yp^3, yp^2, yp, 1, xp, |den|]
// Row/column selection is done with per-lane one-hot FMA weights (branchless).
__global__ void k_accum(const float* __restrict__ hp, const int* __restrict__ lab,
                        float* __restrict__ ws) {
    __shared__ float sacc[NLANES * ACC_STRIDE];   // 120 floats
    const int tid = threadIdx.x;
    for (int i = tid; i < NLANES * ACC_STRIDE; i += 256) sacc[i] = 0.0f;
    __syncthreads();

    const int blk   = blockIdx.x;
    const int b     = blk >> 5;            // 32 blocks per batch
    const int base  = (blk & 31) * 4096;   // pixels per block
    const int wave  = tid >> 5;
    const int lane  = tid & 31;
    const int wbase = base + wave * 512;   // 512 pixels per wave

    const float p0 = hp[b * 6 + 0], p1 = hp[b * 6 + 1], p2 = hp[b * 6 + 2];
    const float p3 = hp[b * 6 + 3], p4 = hp[b * 6 + 4], p5 = hp[b * 6 + 5];

    v8f acc_lo = {};   // rows (l=0..3, i=0..3) -> lane labels 1..4
    v8f acc_hi = {};   // rows 0..3 = (l=4, i)  -> lane label 5

    const int mlo   = lane & 15;
    const int i_lo  = mlo & 3;            // degree row for acc_lo
    const int labA  = (mlo >> 2) + 1;     // lane label 1..4 for acc_lo
    const int j_col = lane & 15;          // B column
    const int kbase = (lane < 16) ? 0 : 2;

    // per-lane one-hot weights (selection without branches)
    const float wA0 = (i_lo == 0) ? 1.0f : 0.0f;
    const float wA1 = (i_lo == 1) ? 1.0f : 0.0f;
    const float wA2 = (i_lo == 2) ? 1.0f : 0.0f;
    const float wA3 = (i_lo == 3) ? 1.0f : 0.0f;

    const float wH0 = (mlo == 0) ? 1.0f : 0.0f;   // acc_hi rows: degree = mlo (rows >=4 all-zero)
    const float wH1 = (mlo == 1) ? 1.0f : 0.0f;
    const float wH2 = (mlo == 2) ? 1.0f : 0.0f;
    const float wH3 = (mlo == 3) ? 1.0f : 0.0f;

    const float wB0 = (j_col == 0) ? 1.0f : 0.0f;
    const float wB1 = (j_col == 1) ? 1.0f : 0.0f;
    const float wB2 = (j_col == 2) ? 1.0f : 0.0f;
    const float wB3 = (j_col == 3) ? 1.0f : 0.0f;
    const float wB4 = (j_col == 4) ? 1.0f : 0.0f;
    const float wB5 = (j_col == 5) ? 1.0f : 0.0f;

    for (int it = 0; it < 16; ++it) {
        const int n = wbase + it * 32 + lane;
        const float fx = (float)(n & (WW - 1));
        const float fy = (float)(n >> 9);
        const float Pp0 = p0 * fx + p1 * fy + p2;
        const float Pp1 = p3 * fy + p4;
        float den = p5 * fy + 1.0f;
        den = (fabsf(den) < EPS_DEN) ? EPS_DEN : den;
        const float xp   = Pp0 / den;
        const float yp   = Pp1 / den;
        const float aden = fabsf(den);
        const int   lb   = lab[b * NPIX + n];

#pragma unroll
        for (int c = 0; c < 8; ++c) {
            const int t0 = c * 4 + kbase;
            const float yp0 = __shfl(yp, t0, 32),   yp1 = __shfl(yp, t0 + 1, 32);
            const float xp0 = __shfl(xp, t0, 32),   xp1 = __shfl(xp, t0 + 1, 32);
            const float ad0 = __shfl(aden, t0, 32), ad1 = __shfl(aden, t0 + 1, 32);
            const int   lb0 = __shfl(lb, t0, 32),   lb1 = __shfl(lb, t0 + 1, 32);

            const float y20 = yp0 * yp0, y30 = y20 * yp0;
            const float y21 = yp1 * yp1, y31 = y21 * yp1;

            const float powA0 = fmaf(wA0, y30, fmaf(wA1, y20, fmaf(wA2, yp0, wA3)));
            const float powA1 = fmaf(wA0, y31, fmaf(wA1, y21, fmaf(wA2, yp1, wA3)));
            const float powH0 = fmaf(wH0, y30, fmaf(wH1, y20, fmaf(wH2, yp0, wH3)));
            const float powH1 = fmaf(wH0, y31, fmaf(wH1, y21, fmaf(wH2, yp1, wH3)));

            const float bv0 = fmaf(wB5, ad0, fmaf(wB4, xp0,
                              fmaf(wB0, y30, fmaf(wB1, y20, fmaf(wB2, yp0, wB3)))));
            const float bv1 = fmaf(wB5, ad1, fmaf(wB4, xp1,
                              fmaf(wB0, y31, fmaf(wB1, y21, fmaf(wB2, yp1, wB3)))));

            v2f a_lo, a_hi, bb;
            a_lo.x = (lb0 == labA) ? powA0 : 0.0f;
            a_lo.y = (lb1 == labA) ? powA1 : 0.0f;
            a_hi.x = (lb0 == 5) ? powH0 : 0.0f;
            a_hi.y = (lb1 == 5) ? powH1 : 0.0f;
            bb.x = bv0;
            bb.y = bv1;

            acc_lo = __builtin_amdgcn_wmma_f32_16x16x4_f32(
                false, a_lo, false, bb, (short)0, acc_lo, false, false);
            acc_hi = __builtin_amdgcn_wmma_f32_16x16x4_f32(
                false, a_hi, false, bb, (short)0, acc_hi, false, false);
        }
    }

    // extract D -> shared accumulators (ds_add_f32)
    const int j = lane & 15;
    if (j < 6) {
#pragma unroll
        for (int e = 0; e < 8; ++e) {
            const int M = (lane < 16) ? e : e + 8;
            atomicAdd(&sacc[(M >> 2) * ACC_STRIDE + (M & 3) * 6 + j], acc_lo[e]);
            if (M < 4)
                atomicAdd(&sacc[4 * ACC_STRIDE + M * 6 + j], acc_hi[e]);
        }
    }
    __syncthreads();
    for (int i = tid; i < NLANES * ACC_STRIDE; i += 256)
        atomicAdd(&ws[ACC_OFF + b * NLANES * ACC_STRIDE + i], sacc[i]);
}

// Pass 2: forty 4x4 ridge solves (partial-pivot Gaussian elimination).
__global__ void k_solve(float* __restrict__ ws) {
    const int t = blockIdx.x * blockDim.x + threadIdx.x;
    if (t >= NB * NLANES) return;
    const float* acc = ws + ACC_OFF + t * ACC_STRIDE;
    float A[4][5];
    for (int i = 0; i < 4; ++i) {
        for (int j = 0; j < 4; ++j) A[i][j] = acc[i * 6 + j] + ((i == j) ? RIDGE : 0.0f);
        A[i][4] = acc[i * 6 + 4];
    }
    for (int k = 0; k < 4; ++k) {
        int piv = k; float mx = fabsf(A[k][k]);
        for (int r = k + 1; r < 4; ++r) { float v = fabsf(A[r][k]); if (v > mx) { mx = v; piv = r; } }
        if (piv != k)
            for (int c = k; c < 5; ++c) { float tmp = A[k][c]; A[k][c] = A[piv][c]; A[piv][c] = tmp; }
        float d = A[k][k]; d = (d == 0.0f) ? 1e-30f : d;
        for (int r = k + 1; r < 4; ++r) {
            float f = A[r][k] / d;
            for (int c = k; c < 5; ++c) A[r][c] -= f * A[k][c];
        }
    }
    float w[4];
    for (int i = 3; i >= 0; --i) {
        float s = A[i][4];
        for (int c = i + 1; c < 4; ++c) s -= A[i][c] * w[c];
        float d = A[i][i]; d = (d == 0.0f) ? 1e-30f : d;
        w[i] = s / d;
    }
    for (int i = 0; i < 4; ++i) ws[W_OFF + t * 4 + i] = w[i];
}

// Pass 3: residual pass -> per (b,l) sum of m*(xp - Y.w)^2
__global__ void k_sse(const float* __restrict__ hp, const int* __restrict__ lab,
                      float* __restrict__ ws) {
    __shared__ float ssse[NLANES];
    __shared__ float sw[NLANES * 4];
    const int tid = threadIdx.x;
    const int blk = blockIdx.x;
    const int b    = blk >> 7;             // 128 blocks per batch
    const int base = (blk & 127) * 1024;   // 1024 pixels per block
    if (tid < NLANES) ssse[tid] = 0.0f;
    if (tid < NLANES * 4) sw[tid] = ws[W_OFF + b * NLANES * 4 + tid];
    __syncthreads();

    const float p0 = hp[b * 6 + 0], p1 = hp[b * 6 + 1], p2 = hp[b * 6 + 2];
    const float p3 = hp[b * 6 + 3], p4 = hp[b * 6 + 4], p5 = hp[b * 6 + 5];

    for (int q = 0; q < 4; ++q) {
        const int n = base + q * 256 + tid;
        const int lb = lab[b * NPIX + n];
        if (lb >= 1 && lb <= NLANES) {
            const float fx = (float)(n & (WW - 1));
            const float fy = (float)(n >> 9);
            float den = p5 * fy + 1.0f;
            den = (fabsf(den) < EPS_DEN) ? EPS_DEN : den;
            const float xp = (p0 * fx + p1 * fy + p2) / den;
            const float yp = (p3 * fy + p4) / den;
            const float* w = &sw[(lb - 1) * 4];
            const float y2 = yp * yp, y3 = y2 * yp;
            const float pred = fmaf(w[0], y3, fmaf(w[1], y2, fmaf(w[2], yp, w[3])));
            const float r = xp - pred;
            atomicAdd(&ssse[lb - 1], r * r);
        }
    }
    __syncthreads();
    if (tid < NLANES)
        atomicAdd(&ws[SSE_OFF + b * NLANES + tid], ssse[tid]);
}

// Pass 4: final scalar
__global__ void k_final(const float* __restrict__ ws, float* __restrict__ out) {
    if (threadIdx.x == 0 && blockIdx.x == 0) {
        float num = 0.0f, nv = 0.0f;
        for (int t = 0; t < NB * NLANES; ++t) {
            const float* acc = ws + ACC_OFF + t * ACC_STRIDE;
            const float cnt    = acc[3 * 6 + 3];   // sum m * 1 * 1
            const float madsum = acc[3 * 6 + 5];   // sum m * |den|
            const float cs  = fmaxf(cnt, 1.0f);
            const float mse = ws[SSE_OFF + t] / cs;
            const float mad = madsum / cs;
            const float valid = (cnt >= 4.0f) ? 1.0f : 0.0f;
            num += valid * mse * mad;
            nv  += valid;
        }
        out[0] = (nv > 0.0f) ? num / fmaxf(nv, 1.0f) : 0.0f;
    }
}

extern "C" void kernel_launch(void* const* d_in, const int* in_sizes, int n_in,
                              void* d_out, int out_size, void* d_ws, size_t ws_size,
                              hipStream_t stream) {
    const float* hp  = (const float*)d_in[0];   // (8,6) f32
    const int*   lab = (const int*)d_in[1];     // (8,256,512) int
    float*       out = (float*)d_out;
    float*       ws  = (float*)d_ws;

    k_zero <<<(WS_FLOATS + 255) / 256, 256, 0, stream>>>(ws);
    k_accum<<<256, 256, 0, stream>>>(hp, lab, ws);
    k_solve<<<1, 64, 0, stream>>>(ws);
    k_sse  <<<1024, 256, 0, stream>>>(hp, lab, ws);
    k_final<<<1, 32, 0, stream>>>(ws, out);
}